// KFLoss_30348238913929
// MI455X (gfx1250) — compile-verified
//
#include <hip/hip_runtime.h>
#include <math.h>

typedef __attribute__((ext_vector_type(2))) float v2f;
typedef __attribute__((ext_vector_type(8))) float v8f;

// Cross-lane sum via the CDNA5 matrix unit:
// D = A x B + 0 with B = all-ones.  A is 16x4 f32: lane m (0-15) supplies
// A[m][0..1] in its two A VGPRs, lane m+16 supplies A[m][2..3].  We put the
// per-lane partial in A VGPR0 and 0 in VGPR1, so row-sum(m) = p[m] + p[m+16].
// D layout: lanes 0-15 VGPR v = D[v][lane] = rowsum(v); lanes 16-31 VGPR v =
// rowsum(v+8).  Summing the 8 D VGPRs in-lane gives:
//   lanes 0-15 : sum of partials of lanes {0..7, 16..23}
//   lanes 16-31: sum of partials of lanes {8..15, 24..31}
// -> lane 0 and lane 16 together hold the full 32-lane sum.
__device__ __forceinline__ float half_wave_sum_wmma(float p) {
    v2f a; a[0] = p;    a[1] = 0.0f;
    v2f b; b[0] = 1.0f; b[1] = 1.0f;
    v8f c = {};
    v8f d = __builtin_amdgcn_wmma_f32_16x16x4_f32(
        /*neg_a=*/false, a, /*neg_b=*/false, b,
        /*c_mod=*/(short)0, c, /*reuse_a=*/false, /*reuse_b=*/false);
    return d[0] + d[1] + d[2] + d[3] + d[4] + d[5] + d[6] + d[7];
}

__global__ void kf_init_acc(double* __restrict__ acc) {
    if (threadIdx.x < 3) acc[threadIdx.x] = 0.0;
}

__global__ void kf_main(const float* __restrict__ pred,
                        const float* __restrict__ target,
                        float* __restrict__ out_kfiou,   // N elements (KFIoU*3)
                        double* __restrict__ acc,        // 3 f64 accumulators
                        int n) {
    __shared__ float sacc[3];
    if (threadIdx.x < 3) sacc[threadIdx.x] = 0.0f;
    __syncthreads();

    float pl = 0.0f, px = 0.0f, pk = 0.0f;   // partial sums: loss, xy_loss, kf_loss

    const int stride = gridDim.x * blockDim.x;
    for (int i = blockIdx.x * blockDim.x + threadIdx.x; i < n; i += stride) {
        const float* pp = pred   + 5 * (size_t)i;
        const float* tt = target + 5 * (size_t)i;
        float xp = pp[0], yp = pp[1], wp = pp[2], hp = pp[3], rp = pp[4];
        float xt = tt[0], yt = tt[1], wt = tt[2], ht = tt[3], rt = tt[4];

        // clip wh to [1e-7, 1e7]
        wp = fminf(fmaxf(wp, 1e-7f), 1e7f);
        hp = fminf(fmaxf(hp, 1e-7f), 1e7f);
        wt = fminf(fmaxf(wt, 1e-7f), 1e7f);
        ht = fminf(fmaxf(ht, 1e-7f), 1e7f);

        // covariance of pred
        float cp = cosf(rp), sp = sinf(rp);
        float swp = 0.25f * wp * wp, shp = 0.25f * hp * hp;
        float a1 = cp * cp * swp + sp * sp * shp;
        float b1 = cp * sp * (swp - shp);
        float c1 = sp * sp * swp + cp * cp * shp;

        // covariance of target
        float ct = cosf(rt), st = sinf(rt);
        float swt = 0.25f * wt * wt, sht = 0.25f * ht * ht;
        float a2 = ct * ct * swt + st * st * sht;
        float b2 = ct * st * (swt - sht);
        float c2 = st * st * swt + ct * ct * sht;

        // smooth-L1 on centers (BETA = 1)
        float dx = fabsf(xp - xt), dy = fabsf(yp - yt);
        float lx = (dx < 1.0f) ? 0.5f * dx * dx : dx - 0.5f;
        float ly = (dy < 1.0f) ? 0.5f * dy * dy : dy - 0.5f;
        float xyl = lx + ly;

        float vbp = wp * hp;
        float vbt = wt * ht;

        // inv(Sp + St)
        float A = a1 + a2, B = b1 + b2, C = c1 + c2;
        float detm = A * C - B * B;
        float i00 =  C / detm, i01 = -B / detm, i11 = A / detm;  // i10 == i01

        // K = Sp @ inv_sum
        float k00 = a1 * i00 + b1 * i01;
        float k01 = a1 * i01 + b1 * i11;
        float k10 = b1 * i00 + c1 * i01;
        float k11 = b1 * i01 + c1 * i11;

        // Sigma = Sp - K @ Sp
        float s00 = a1 - (k00 * a1 + k01 * b1);
        float s01 = b1 - (k00 * b1 + k01 * c1);
        float s10 = b1 - (k10 * a1 + k11 * b1);
        float s11 = c1 - (k10 * b1 + k11 * c1);

        float det2 = s00 * s11 - s01 * s10;
        float vb   = (det2 > 0.0f) ? 4.0f * sqrtf(det2) : 0.0f;

        float kfiou = vb / (vbp + vbt - vb + 1e-6f);
        float kfl   = 1.0f - kfiou;
        float loss  = fmaxf(xyl + kfl, 0.0f);

        out_kfiou[i] = kfiou * 3.0f;

        pl += loss;
        px += xyl;
        pk += kfl;
    }

    // Wave reduction on the matrix unit (EXEC is all-1s here: no early returns,
    // fully reconverged after the grid-stride loop).
    float tl = half_wave_sum_wmma(pl);
    float tx = half_wave_sum_wmma(px);
    float tk = half_wave_sum_wmma(pk);

    int lane = threadIdx.x & 31;
    if (lane == 0 || lane == 16) {   // the two complementary half-wave sums
        atomicAdd(&sacc[0], tl);
        atomicAdd(&sacc[1], tx);
        atomicAdd(&sacc[2], tk);
    }
    __syncthreads();

    if (threadIdx.x == 0) {
        atomicAdd(&acc[0], (double)sacc[0]);
        atomicAdd(&acc[1], (double)sacc[1]);
        atomicAdd(&acc[2], (double)sacc[2]);
    }
}

__global__ void kf_finalize(const double* __restrict__ acc,
                            float* __restrict__ out, int n) {
    if (threadIdx.x < 3) {
        out[threadIdx.x] = (float)(acc[threadIdx.x] / (double)n);
    }
}

extern "C" void kernel_launch(void* const* d_in, const int* in_sizes, int n_in,
                              void* d_out, int out_size, void* d_ws, size_t ws_size,
                              hipStream_t stream) {
    const float* pred   = (const float*)d_in[0];
    const float* target = (const float*)d_in[1];
    int n = in_sizes[0] / 5;

    float*  out = (float*)d_out;       // [loss_mean, xy_mean, kf_mean, KFIoU*3...]
    double* acc = (double*)d_ws;       // 3 f64 accumulators

    kf_init_acc<<<1, 32, 0, stream>>>(acc);

    const int block = 256;             // 8 wave32 waves
    int grid = (n + block - 1) / block;
    if (grid > 2048) grid = 2048;      // grid-stride: ~4 rows/thread, few atomics

    kf_main<<<grid, block, 0, stream>>>(pred, target, out + 3, acc, n);
    kf_finalize<<<1, 32, 0, stream>>>(acc, out, n);
}